// AVWGCN2_3470333575183
// MI455X (gfx1250) — compile-verified
//
#include <hip/hip_runtime.h>
#include <math.h>

// Problem constants (from the reference)
#define Bn    32        // batch
#define Nn    8192      // nodes
#define CIN   32        // dim_in
#define COUT  64        // dim_out
#define ED    16        // embed_dim
#define ROWS_PER_WG 32  // softmax/output rows per workgroup
#define KC    16        // m-chunk per K-loop step (double buffered)

#define XS_STRIDE  1032 // floats; bank shift 8/row -> conflict-free B-tile loads
#define ACC_STRIDE 1028 // floats; halves land 32 banks apart on dump

typedef float v2f __attribute__((ext_vector_type(2)));
typedef float v8f __attribute__((ext_vector_type(8)));

// ---- dynamic LDS layout (floats) ----
//  [0, 33024)  : xsA @0 (16512) | xsB @16512 (16512)
//                stats phase aliases xsA: echunk 4096 @0, red_m @4096, red_s @4352
//                epilogue aliases both: accs 32 x ACC_STRIDE (32896) @0
#define OFF_XSA   0
#define OFF_XSB   16512
#define OFF_ESA   33024   // 16x16 E rows, buffer A
#define OFF_ESB   33280   // buffer B
#define OFF_ER    33536   // 32x16 embedding rows of this block
#define OFF_REDM  4096
#define OFF_REDS  4352
#define OFF_SMX   34048   // 32 row maxima
#define OFF_SINV  34080   // 32 reciprocal softmax sums
#define OFF_W     34112   // 32x64 per-node weight matrix (combine phase)
#define OFF_BIAS  36160   // 64 per-node bias
#define SMEM_FLOATS 36224 // 144896 bytes (two blocks fit in a 320KB WGP)

// 32 consecutive floats (8 x b128), global -> LDS, ASYNCcnt-tracked.
// Per ISA 10.4/15.18: INST_OFFSET is added to BOTH the LDS and global address.
__device__ __forceinline__ void async_copy_row32(unsigned lds, const float* g) {
    asm volatile(
        "global_load_async_to_lds_b128 %0, %1, off\n\t"
        "global_load_async_to_lds_b128 %0, %1, off offset:16\n\t"
        "global_load_async_to_lds_b128 %0, %1, off offset:32\n\t"
        "global_load_async_to_lds_b128 %0, %1, off offset:48\n\t"
        "global_load_async_to_lds_b128 %0, %1, off offset:64\n\t"
        "global_load_async_to_lds_b128 %0, %1, off offset:80\n\t"
        "global_load_async_to_lds_b128 %0, %1, off offset:96\n\t"
        "global_load_async_to_lds_b128 %0, %1, off offset:112"
        :: "v"(lds), "v"(g) : "memory");
}

__device__ __forceinline__ void async_wait0() {
    asm volatile("s_wait_asynccnt 0" ::: "memory");
}

// Prefetch one K-chunk: xs[k][col] (k = m-ms, col = b*32+c) plus 16 E rows.
__device__ __forceinline__ void stage_chunk_async(float* xsbuf, float* esbuf,
                                                  const float* x, const float* E,
                                                  int ms, int t) {
    const int k  = t >> 4;           // 0..15 : m-row within chunk
    const int cb = (t & 15) * 64;    // 64 cols = 2 batches x 32 channels
    #pragma unroll
    for (int seg = 0; seg < 2; ++seg) {
        const int b = (cb >> 5) + seg;
        const float*   g = x + (size_t)b * (Nn * CIN) + (size_t)(ms + k) * CIN;
        const unsigned l = (unsigned)(uintptr_t)(xsbuf + k * XS_STRIDE + cb + seg * CIN);
        async_copy_row32(l, g);
    }
    if (t < 64) {   // waves 0-1 only (wave-uniform branch): 16x16 E rows
        const unsigned l = (unsigned)(uintptr_t)(esbuf + t * 4);
        const float*   g = E + (size_t)ms * ED + t * 4;
        asm volatile("global_load_async_to_lds_b128 %0, %1, off"
                     :: "v"(l), "v"(g) : "memory");
    }
}

extern "C" __global__ void __launch_bounds__(256)
avwgcn_fused(const float* __restrict__ x,    // [B, N, CIN]
             const float* __restrict__ E,    // [N, ED]
             const float* __restrict__ Wp,   // [ED, CIN, COUT]
             const float* __restrict__ bp,   // [ED, COUT]
             float* __restrict__ out)        // [B, N, COUT]
{
    extern __shared__ float S[];
    const int t    = threadIdx.x;
    const int wave = t >> 5;
    const int lane = t & 31;
    const int half = lane >> 4;   // K-half for WMMA A/B layout
    const int lrow = lane & 15;   // M (or N) index within a 16-tile
    const int n0   = blockIdx.x * ROWS_PER_WG;

    float* xsA  = S + OFF_XSA;
    float* xsB  = S + OFF_XSB;
    float* esA  = S + OFF_ESA;
    float* esB  = S + OFF_ESB;
    float* er   = S + OFF_ER;
    float* smx  = S + OFF_SMX;
    float* sinv = S + OFF_SINV;

    // stage this block's 32 embedding rows (32x16 floats)
    er[t]       = E[n0 * ED + t];
    er[t + 256] = E[n0 * ED + 256 + t];

    // kick off async prefetch of K-chunk 0 into buffer B; it lands while the
    // softmax-statistics pass below runs out of the (disjoint) xsA region.
    stage_chunk_async(xsB, esB, x, E, 0, t);
    __syncthreads();

    // ---------------- pass 1: softmax row statistics ----------------
    // row r handled by threads {r*8 .. r*8+7}; online max/sum over m.
    {
        float* ech  = S;               // 256 x 16 staged E chunk (aliases xsA)
        float* redm = S + OFF_REDM;
        float* reds = S + OFF_REDS;
        const int row = t >> 3;
        const int j   = t & 7;
        float mx = 0.0f, sm = 0.0f;    // valid start: relu scores are >= 0
        for (int ms = 0; ms < Nn; ms += 256) {
            __syncthreads();
            {   // thread t stages E row (ms+t): 16 floats = 4 float4
                const float4* src = (const float4*)(E + (size_t)ms * ED);
                float4*       dst = (float4*)ech;
                #pragma unroll
                for (int q = 0; q < 4; ++q) dst[t * 4 + q] = src[t * 4 + q];
            }
            __syncthreads();
            for (int mm = j; mm < 256; mm += 8) {
                float s = 0.0f;
                #pragma unroll
                for (int d = 0; d < ED; ++d) s += er[row * ED + d] * ech[mm * ED + d];
                s = fmaxf(s, 0.0f);                      // relu
                if (s > mx) { sm = sm * __expf(mx - s) + 1.0f; mx = s; }
                else        { sm += __expf(s - mx); }
            }
        }
        __syncthreads();
        redm[t] = mx; reds[t] = sm;
        __syncthreads();
        if (t < ROWS_PER_WG) {         // merge the 8 partials of row t
            float m0 = redm[t * 8], s0 = reds[t * 8];
            #pragma unroll
            for (int jj = 1; jj < 8; ++jj) {
                float m1 = redm[t * 8 + jj], s1 = reds[t * 8 + jj];
                if (m1 > m0) { s0 = s0 * __expf(m0 - m1) + s1; m0 = m1; }
                else         { s0 += s1 * __expf(m1 - m0); }
            }
            smx[t] = m0; sinv[t] = 1.0f / s0;
        }
        __syncthreads();
    }

    // per-lane registers for on-the-fly A-tile generation
    float erA[ED], erB[ED];
    #pragma unroll
    for (int d = 0; d < ED; ++d) {
        erA[d] = er[lrow * ED + d];          // row tile 0: rows n0 + 0..15
        erB[d] = er[(16 + lrow) * ED + d];   // row tile 1: rows n0 + 16..31
    }
    const float mxA = smx[lrow],      invA = sinv[lrow];
    const float mxB = smx[16 + lrow], invB = sinv[16 + lrow];

    v8f acc0[8], acc1[8];
    {
        v8f z = {0.f, 0.f, 0.f, 0.f, 0.f, 0.f, 0.f, 0.f};
        #pragma unroll
        for (int ct = 0; ct < 8; ++ct) { acc0[ct] = z; acc1[ct] = z; }
    }

    // chunk 0 (issued before stats) must be resident before first use
    async_wait0();
    __syncthreads();

    // ------- pass 2: fused softmax(A) x X GEMM via f32 WMMA, async double-buffer -------
    const int NCH = Nn / KC;   // 512 chunks
    for (int ci = 0; ci < NCH; ++ci) {
        float* cur  = (ci & 1) ? xsA : xsB;   // chunk 0 is in buffer B
        float* curE = (ci & 1) ? esA : esB;
        if (ci + 1 < NCH)                     // prefetch next chunk while computing
            stage_chunk_async((ci & 1) ? xsB : xsA, (ci & 1) ? esB : esA,
                              x, E, (ci + 1) * KC, t);

        #pragma unroll 1
        for (int kk = 0; kk < 4; ++kk) {
            // A layout (16x4 f32): lanes 0-15 hold K=0,1; lanes 16-31 hold K=2,3
            const int k0 = kk * 4 + half * 2;
            float sx0 = 0.f, sy0 = 0.f, sx1 = 0.f, sy1 = 0.f;
            #pragma unroll
            for (int d = 0; d < ED; ++d) {
                const float e0 = curE[k0 * ED + d];
                const float e1 = curE[k0 * ED + ED + d];
                sx0 += erA[d] * e0;  sy0 += erA[d] * e1;
                sx1 += erB[d] * e0;  sy1 += erB[d] * e1;
            }
            v2f a0, a1;
            a0.x = __expf(fmaxf(sx0, 0.f) - mxA) * invA;
            a0.y = __expf(fmaxf(sy0, 0.f) - mxA) * invA;
            a1.x = __expf(fmaxf(sx1, 0.f) - mxB) * invB;
            a1.y = __expf(fmaxf(sy1, 0.f) - mxB) * invB;

            #pragma unroll
            for (int ct = 0; ct < 8; ++ct) {
                const int col = wave * 128 + ct * 16 + lrow;
                v2f bv;   // B layout (4x16): lanes 0-15 K=0/1, lanes 16-31 K=2/3
                bv.x = cur[(kk * 4 + half * 2    ) * XS_STRIDE + col];
                bv.y = cur[(kk * 4 + half * 2 + 1) * XS_STRIDE + col];
                acc0[ct] = __builtin_amdgcn_wmma_f32_16x16x4_f32(
                    false, a0, false, bv, (short)0, acc0[ct], false, false);
                acc1[ct] = __builtin_amdgcn_wmma_f32_16x16x4_f32(
                    false, a1, false, bv, (short)0, acc1[ct], false, false);
            }
        }
        async_wait0();     // own prefetches landed in LDS
        __syncthreads();   // all waves done reading `cur`, all writes visible
    }

    // ---------------- epilogue: dump x_g (+ identity term) into LDS ----------------
    float* accs = S;            // 32 x ACC_STRIDE (aliases both xs buffers)
    #pragma unroll
    for (int ct = 0; ct < 8; ++ct) {
        const int col = wave * 128 + ct * 16 + lrow;
        const int b = col >> 5, c = col & 31;
        const float* xcol = x + (size_t)b * (Nn * CIN) + c;
        #pragma unroll
        for (int g = 0; g < 8; ++g) {   // D layout: lanes 0-15 M=g, lanes 16-31 M=g+8
            const int nl0 = g + half * 8;
            const int nl1 = 16 + g + half * 8;
            accs[nl0 * ACC_STRIDE + col] = acc0[ct][g] + xcol[(size_t)(n0 + nl0) * CIN];
            accs[nl1 * ACC_STRIDE + col] = acc1[ct][g] + xcol[(size_t)(n0 + nl1) * CIN];
        }
    }
    __syncthreads();

    // ---------------- per-node combine: out = x_g @ W(n) + bias(n) ----------------
    float* wm = S + OFF_W;      // 32 x 64 : wm[i*64 + o]
    float* bb = S + OFF_BIAS;   // 64
    for (int nl = 0; nl < ROWS_PER_WG; ++nl) {
        #pragma unroll
        for (int e = 0; e < 8; ++e) {           // generate W(n): 2048 elems / 256 thr
            const int io = t + e * 256;
            float v = 0.0f;
            #pragma unroll
            for (int d = 0; d < ED; ++d) v += er[nl * ED + d] * Wp[d * (CIN * COUT) + io];
            wm[io] = v;
        }
        if (t < COUT) {
            float v = 0.0f;
            #pragma unroll
            for (int d = 0; d < ED; ++d) v += er[nl * ED + d] * bp[d * COUT + t];
            bb[t] = v;
        }
        __syncthreads();
        #pragma unroll
        for (int e = 0; e < 8; ++e) {           // 32 b x 64 o outputs for this node
            const int idx = t + e * 256;
            const int b = idx >> 6, o = idx & 63;
            float v = bb[o];
            #pragma unroll
            for (int i = 0; i < CIN; ++i)
                v += accs[nl * ACC_STRIDE + b * CIN + i] * wm[i * COUT + o];
            out[(size_t)b * (Nn * COUT) + (size_t)(n0 + nl) * COUT + o] = v;
        }
        __syncthreads();
    }
}

extern "C" void kernel_launch(void* const* d_in, const int* in_sizes, int n_in,
                              void* d_out, int out_size, void* d_ws, size_t ws_size,
                              hipStream_t stream) {
    const float* x  = (const float*)d_in[0];   // [32, 8192, 32]
    const float* E  = (const float*)d_in[1];   // [8192, 16]
    const float* Wp = (const float*)d_in[2];   // [16, 32, 64]
    const float* bp = (const float*)d_in[3];   // [16, 64]
    float* out = (float*)d_out;                // [32, 8192, 64]

    const size_t smem = SMEM_FLOATS * sizeof(float);   // ~141.5 KB
    (void)hipFuncSetAttribute((const void*)avwgcn_fused,
                              hipFuncAttributeMaxDynamicSharedMemorySize, (int)smem);
    dim3 grid(Nn / ROWS_PER_WG);   // 256 workgroups
    dim3 block(256);               // 8 waves (wave32)
    avwgcn_fused<<<grid, block, smem, stream>>>(x, E, Wp, bp, out);
}